// SetBatchNorm_8581344657855
// MI455X (gfx1250) — compile-verified
//
#include <hip/hip_runtime.h>

// Masked batch-norm over axis N (B=256, N=512, D=256), f32.
// HBM-bound: 128MB in + 128MB out; second read of x designed to hit the 192MB L2.
// CDNA5 paths used: global_load_async_to_lds_b32 (+ s_wait_asynccnt),
// global_prefetch_b8, non-temporal (TH) loads/stores.

#define BN_B 256
#define BN_N 512
#define BN_D 256
#define TPB  512              // 16 waves (wave32)
#define NROW (TPB / 64)       // 8 row-groups; 64 column-groups of 4 floats

typedef float v4f __attribute__((ext_vector_type(4)));   // clang native vector:
                                                         // valid for NT builtins

__global__ __launch_bounds__(TPB) void masked_bn_kernel(
    const float* __restrict__ x,
    const unsigned char* __restrict__ mask,   // jnp.bool_: 1 byte/elem, nonzero = dead
    const float* __restrict__ wgt,
    const float* __restrict__ bia,
    float* __restrict__ out)
{
    __shared__ unsigned char mrow[BN_N];      // raw mask row (async DMA target)
    __shared__ float aliveF[BN_N];            // 0.0 / 1.0 alive flags
    __shared__ v4f redS [NROW][64];           // partial sums
    __shared__ v4f redS2[NROW][64];           // partial sums of squares

    const int b   = blockIdx.x;
    const int tid = threadIdx.x;
    const int c   = tid & 63;                 // column group: d = 4*c .. 4*c+3
    const int r   = tid >> 6;                 // row group: n = r, r+8, ...

    const float* xb = x   + (size_t)b * (BN_N * BN_D);
    float*       ob = out + (size_t)b * (BN_N * BN_D);

    // ---- async DMA: mask row (512B) global -> LDS, tracked by ASYNCcnt ----
    if (tid < BN_N / 4) {
        unsigned lds_off = (unsigned)(size_t)(&mrow[0]) + (unsigned)(tid * 4);
        const unsigned char* g = mask + (size_t)b * BN_N + (size_t)(tid * 4);
        asm volatile("global_load_async_to_lds_b32 %0, %1, off"
                     :: "v"(lds_off), "v"(g)
                     : "memory");
    }
    asm volatile("s_wait_asynccnt 0x0" ::: "memory");
    __syncthreads();

    // ---- convert to float alive flags ----
    for (int i = tid; i < BN_N; i += TPB)
        aliveF[i] = mrow[i] ? 0.0f : 1.0f;
    __syncthreads();

    // ---- per-batch alive count (broadcast LDS reads; identical in all threads) ----
    float cnt = 0.0f;
    {
        const v4f* a4 = (const v4f*)aliveF;
        #pragma unroll 8
        for (int i = 0; i < BN_N / 4; ++i) {
            v4f a = a4[i];
            cnt += (a[0] + a[1]) + (a[2] + a[3]);
        }
    }

    // ---- pass 1: masked sum and sum-of-squares over N (coalesced b128) ----
    v4f s  = (v4f)(0.0f);
    v4f s2 = (v4f)(0.0f);
    const float* colp = xb + 4 * c;
    #pragma unroll 4
    for (int n = r; n < BN_N; n += NROW) {
        const float* p = colp + (size_t)n * BN_D;
        if (n + 4 * NROW < BN_N)
            __builtin_prefetch(p + 4 * NROW * BN_D, 0, 0);   // global_prefetch_b8
        v4f v = *(const v4f*)p;
        float a = aliveF[n];
        v *= a;
        s  += v;
        s2 += v * v;
    }
    redS [r][c] = s;
    redS2[r][c] = s2;
    __syncthreads();

    v4f S  = (v4f)(0.0f);
    v4f S2 = (v4f)(0.0f);
    #pragma unroll
    for (int rr = 0; rr < NROW; ++rr) {
        S  += redS [rr][c];
        S2 += redS2[rr][c];
    }

    // ---- stats: mean, var = (S2 - 2*mean*S + N*mean^2)/cnt, std ----
    const float cntc = fmaxf(cnt, 1.0f);
    const bool  bm   = cntc > 1.0f;          // batch_mask: >1 valid elements
    const float inv  = 1.0f / cntc;
    const float Nf   = (float)BN_N;

    v4f mean = bm ? (S * inv) : S;
    v4f var  = (S2 - 2.0f * mean * S + Nf * mean * mean) * inv;

    v4f stdv;
    stdv[0] = bm ? sqrtf(var[0] + 1e-6f) : 1.0f;
    stdv[1] = bm ? sqrtf(var[1] + 1e-6f) : 1.0f;
    stdv[2] = bm ? sqrtf(var[2] + 1e-6f) : 1.0f;
    stdv[3] = bm ? sqrtf(var[3] + 1e-6f) : 1.0f;

    const v4f w4 = *(const v4f*)(wgt + 4 * c);
    const v4f b4 = *(const v4f*)(bia + 4 * c);
    v4f scale = w4 / stdv;
    v4f shift = b4 - mean * scale;

    // ---- pass 2: re-read x (L2-hot, last use -> NT load), write out NT ----
    #pragma unroll 4
    for (int n = r; n < BN_N; n += NROW) {
        const v4f* p = (const v4f*)(colp + (size_t)n * BN_D);
        v4f v = __builtin_nontemporal_load(p);
        float a = aliveF[n];
        v4f o;
        o[0] = fmaf(v[0] * a, scale[0], shift[0]);
        o[1] = fmaf(v[1] * a, scale[1], shift[1]);
        o[2] = fmaf(v[2] * a, scale[2], shift[2]);
        o[3] = fmaf(v[3] * a, scale[3], shift[3]);
        __builtin_nontemporal_store(o, (v4f*)(ob + (size_t)n * BN_D + 4 * c));
    }
}

extern "C" void kernel_launch(void* const* d_in, const int* in_sizes, int n_in,
                              void* d_out, int out_size, void* d_ws, size_t ws_size,
                              hipStream_t stream) {
    const float*         x    = (const float*)d_in[0];
    const unsigned char* mask = (const unsigned char*)d_in[1]; // bool array: 1B/elem
    const float*         wgt  = (const float*)d_in[2];
    const float*         bia  = (const float*)d_in[3];
    float*               out  = (float*)d_out;

    dim3 grid(BN_B), block(TPB);
    masked_bn_kernel<<<grid, block, 0, stream>>>(x, mask, wgt, bia, out);
}